// Part_Graph_20813411517052
// MI455X (gfx1250) — compile-verified
//
#include <hip/hip_runtime.h>
#include <math.h>

#define HW 16384
#define NB 4

typedef float v2f __attribute__((ext_vector_type(2)));
typedef float v8f __attribute__((ext_vector_type(8)));

__device__ __forceinline__ v8f wmma4(v2f a, v2f b, v8f c) {
  // D = A(16x4 f32) * B(4x16 f32) + C(16x16 f32)
  return __builtin_amdgcn_wmma_f32_16x16x4_f32(false, a, false, b, (short)0, c, false, false);
}

__device__ __forceinline__ float sigm(float x) { return 1.0f / (1.0f + __expf(-x)); }
__device__ __forceinline__ int clampi(int v, int lo, int hi) {
  return v < lo ? lo : (v > hi ? hi : v);
}

// ---------------------------------------------------------------------------
// K1: decomposition maps + softmax, contexture attention (sigmoid), PA scalars,
//     and copy p_nodes into Zg channels 10..19.  One thread = one pixel.
// ---------------------------------------------------------------------------
__global__ __launch_bounds__(256) void k1_attention(
    const float* __restrict__ h_nodes, const float* __restrict__ p_nodes,
    const float* __restrict__ W_dau, const float* __restrict__ b_dau,
    const float* __restrict__ W_dal, const float* __restrict__ b_dal,
    const float* __restrict__ Wa0, const float* __restrict__ Wa1,
    const float* __restrict__ Wa2, const float* __restrict__ Wa3,
    const float* __restrict__ Wa4, const float* __restrict__ Wa5,
    const float* __restrict__ b_att,
    float* __restrict__ out_dmu, float* __restrict__ out_dml,
    float* __restrict__ out_fdep, float* __restrict__ PA, float* __restrict__ Zg)
{
  int p = blockIdx.x * 256 + threadIdx.x;           // [0, 65536)
  int b = p >> 14, hw = p & (HW - 1);

  float h0[10], h1[10], pn[6][10];
#pragma unroll
  for (int c = 0; c < 10; c++) {
    h0[c] = h_nodes[((0 * NB + b) * 10 + c) * HW + hw];
    h1[c] = h_nodes[((1 * NB + b) * 10 + c) * HW + hw];
  }
#pragma unroll
  for (int n = 0; n < 6; n++)
#pragma unroll
    for (int c = 0; c < 10; c++) {
      float v = p_nodes[((n * NB + b) * 10 + c) * HW + hw];
      pn[n][c] = v;
      Zg[((n * NB + b) * 20 + 10 + c) * HW + hw] = v;   // hidden half of gate input
    }

  // decomposition maps + softmax (upper: 5ch from h0, lower: 3ch from h1)
  float mu[5], ml[3];
#pragma unroll
  for (int c = 0; c < 5; c++) {
    float s = b_dau[c];
#pragma unroll
    for (int k = 0; k < 10; k++) s += W_dau[c * 10 + k] * h0[k];
    mu[c] = s;
    out_dmu[(b * 5 + c) * HW + hw] = s;
  }
#pragma unroll
  for (int c = 0; c < 3; c++) {
    float s = b_dal[c];
#pragma unroll
    for (int k = 0; k < 10; k++) s += W_dal[c * 10 + k] * h1[k];
    ml[c] = s;
    out_dml[(b * 3 + c) * HW + hw] = s;
  }
  float mxu = mu[0];
#pragma unroll
  for (int c = 1; c < 5; c++) mxu = fmaxf(mxu, mu[c]);
  float eu[5], seu = 0.0f;
#pragma unroll
  for (int c = 0; c < 5; c++) { eu[c] = __expf(mu[c] - mxu); seu += eu[c]; }
  float invu = 1.0f / seu;
  float mxl = ml[0];
#pragma unroll
  for (int c = 1; c < 3; c++) mxl = fmaxf(mxl, ml[c]);
  float el[3], sel = 0.0f;
#pragma unroll
  for (int c = 0; c < 3; c++) { el[c] = __expf(ml[c] - mxl); sel += el[c]; }
  float invl = 1.0f / sel;

  // parent attention scalar per decomposition unit
  PA[(0 * NB + b) * HW + hw] = eu[1] * invu;
  PA[(1 * NB + b) * HW + hw] = eu[2] * invu;
  PA[(2 * NB + b) * HW + hw] = eu[3] * invu;
  PA[(3 * NB + b) * HW + hw] = eu[4] * invu;
  PA[(4 * NB + b) * HW + hw] = el[1] * invl;
  PA[(5 * NB + b) * HW + hw] = el[2] * invl;

  // contexture attention: PART_LIST_LIST = [[1],[0,2,3,4],[1],[1],[1,5],[4]]
  float f[6];
  {
    float s = 0.0f;
#pragma unroll
    for (int k = 0; k < 10; k++) s += Wa0[k] * pn[1][k];
    f[0] = s;
  }
  {
    float s = 0.0f;
#pragma unroll
    for (int k = 0; k < 10; k++) s += Wa1[k] * pn[0][k];
#pragma unroll
    for (int k = 0; k < 10; k++) s += Wa1[10 + k] * pn[2][k];
#pragma unroll
    for (int k = 0; k < 10; k++) s += Wa1[20 + k] * pn[3][k];
#pragma unroll
    for (int k = 0; k < 10; k++) s += Wa1[30 + k] * pn[4][k];
    f[1] = s;
  }
  {
    float s = 0.0f;
#pragma unroll
    for (int k = 0; k < 10; k++) s += Wa2[k] * pn[1][k];
    f[2] = s;
  }
  {
    float s = 0.0f;
#pragma unroll
    for (int k = 0; k < 10; k++) s += Wa3[k] * pn[1][k];
    f[3] = s;
  }
  {
    float s = 0.0f;
#pragma unroll
    for (int k = 0; k < 10; k++) s += Wa4[k] * pn[1][k];
#pragma unroll
    for (int k = 0; k < 10; k++) s += Wa4[10 + k] * pn[5][k];
    f[4] = s;
  }
  {
    float s = 0.0f;
#pragma unroll
    for (int k = 0; k < 10; k++) s += Wa5[k] * pn[4][k];
    f[5] = s;
  }
#pragma unroll
  for (int i = 0; i < 6; i++)
    out_fdep[(i * NB + b) * HW + hw] = sigm(f[i] + b_att[i]);
}

// ---------------------------------------------------------------------------
// K2: message = relu(bn(Wdec2 @ relu(bn(Wdec1 @ [a*parent;child])))) +
//               relu((2-att)*(Wproj @ xp)*s + t)
//     One wave = 16 pixels.  All weight tiles zero-padded in LDS so every
//     A-operand / affine read is unconditional (branch-free).
// ---------------------------------------------------------------------------
__global__ __launch_bounds__(128) void k2_message(
    const float* __restrict__ h_nodes, const float* __restrict__ p_nodes,
    const float* __restrict__ xp,
    const float* __restrict__ W_dec1, const float* __restrict__ dec1_s, const float* __restrict__ dec1_t,
    const float* __restrict__ W_dec2, const float* __restrict__ dec2_s, const float* __restrict__ dec2_t,
    const float* __restrict__ W_proj, const float* __restrict__ proj_s, const float* __restrict__ proj_t,
    const float* __restrict__ fdep, const float* __restrict__ PA,
    float* __restrict__ Zg, float* __restrict__ Zc)
{
  __shared__ float sW1[640];      // W_dec1 padded to [32,20]
  __shared__ float sW2[320];      // W_dec2 padded to [16,20]
  __shared__ float sWp[4096];     // W_proj[n] padded to [16,256]
  __shared__ float sS1[32], sT1[32];
  __shared__ float sS2[16], sT2[16], sPs[16], sPt[16];
  __shared__ float hbuf[4][512];  // per-wave 32x16 relay tile (rows 20..31 junk)

  int tid = threadIdx.x;
  int wid = blockIdx.x * 4 + (tid >> 5);  // 4096 waves per node; blocks never straddle n
  int n = wid >> 12;

  for (int i = tid; i < 400; i += 128) sW1[i] = W_dec1[i];
  for (int i = 400 + tid; i < 640; i += 128) sW1[i] = 0.0f;
  for (int i = tid; i < 200; i += 128) sW2[i] = W_dec2[i];
  for (int i = 200 + tid; i < 320; i += 128) sW2[i] = 0.0f;
  const float* Wp = W_proj + n * 2560;
  for (int i = tid; i < 2560; i += 128) sWp[i] = Wp[i];
  for (int i = 2560 + tid; i < 4096; i += 128) sWp[i] = 0.0f;
  if (tid < 32) {
    float s = 0.0f, t = 0.0f;
    if (tid < 20) { s = dec1_s[tid]; t = dec1_t[tid]; }
    sS1[tid] = s; sT1[tid] = t;
  }
  if (tid >= 32 && tid < 48) {
    int i = tid - 32;
    float a = 0.0f, bb = 0.0f, c = 0.0f, d = 0.0f;
    if (i < 10) { a = dec2_s[i]; bb = dec2_t[i]; c = proj_s[n * 10 + i]; d = proj_t[n * 10 + i]; }
    sS2[i] = a; sT2[i] = bb; sPs[i] = c; sPt[i] = d;
  }
  __syncthreads();

  int lane = tid & 31;
  int t = wid & 4095;
  int p0 = t * 16;
  int b = p0 >> 14, hw0 = p0 & (HW - 1);
  int col = lane & 15;            // pixel within tile (N) / weight row (M) for A
  int kh = (lane >> 4) << 1;      // K sub-offset per lane half
  int r1 = col + 16;

  int pi = (n < 4) ? 0 : 1;
  const float* par = h_nodes + ((pi * NB + b) * 10) * HW + hw0 + col;
  const float* chd = p_nodes + ((n * NB + b) * 10) * HW + hw0 + col;
  float pa = PA[(n * NB + b) * HW + hw0 + col];

  // ---- stage 1: 20x20 GEMM over x = [pa*parent(10); child(10)] ----
  v8f acc0 = {}, acc1 = {};
#pragma unroll
  for (int kk = 0; kk < 2; kk++) {           // k in [0,8): pure parent
    int k0 = kk * 4 + kh;
    v2f bm, a0, a1;
    bm.x = par[k0 * HW] * pa;
    bm.y = par[(k0 + 1) * HW] * pa;
    a0.x = sW1[col * 20 + k0]; a0.y = sW1[col * 20 + k0 + 1];
    a1.x = sW1[r1 * 20 + k0];  a1.y = sW1[r1 * 20 + k0 + 1];
    acc0 = wmma4(a0, bm, acc0);
    acc1 = wmma4(a1, bm, acc1);
  }
  {                                          // k in [8,12): mixed -> value selects
    int k0 = 8 + kh;                         // 8 (parent) or 10 (child)
    bool isp = (kh == 0);
    const float* src = isp ? par : chd;
    int off = isp ? 8 : 0;
    float sc = isp ? pa : 1.0f;
    v2f bm, a0, a1;
    bm.x = src[off * HW] * sc;
    bm.y = src[(off + 1) * HW] * sc;
    a0.x = sW1[col * 20 + k0]; a0.y = sW1[col * 20 + k0 + 1];
    a1.x = sW1[r1 * 20 + k0];  a1.y = sW1[r1 * 20 + k0 + 1];
    acc0 = wmma4(a0, bm, acc0);
    acc1 = wmma4(a1, bm, acc1);
  }
#pragma unroll
  for (int kk = 3; kk < 5; kk++) {           // k in [12,20): pure child
    int k0 = kk * 4 + kh;
    int c0 = k0 - 10;
    v2f bm, a0, a1;
    bm.x = chd[c0 * HW];
    bm.y = chd[(c0 + 1) * HW];
    a0.x = sW1[col * 20 + k0]; a0.y = sW1[col * 20 + k0 + 1];
    a1.x = sW1[r1 * 20 + k0];  a1.y = sW1[r1 * 20 + k0 + 1];
    acc0 = wmma4(a0, bm, acc0);
    acc1 = wmma4(a1, bm, acc1);
  }

  float* hb = hbuf[tid >> 5];
  int mb = (lane < 16) ? 0 : 8;
#pragma unroll
  for (int g = 0; g < 8; g++) {              // rows 0..15, unconditional
    int M = mb + g;
    hb[M * 16 + col] = fmaxf(acc0[g] * sS1[M] + sT1[M], 0.0f);
  }
#pragma unroll
  for (int g = 0; g < 8; g++) {              // rows 16..31, unconditional (junk rows unread)
    int M = 16 + mb + g;
    hb[M * 16 + col] = fmaxf(acc1[g] * sS1[M] + sT1[M], 0.0f);
  }
  __syncthreads();

  // ---- stage 2: 10x20 GEMM on relayed h ----
  v8f acc2 = {};
#pragma unroll
  for (int kk = 0; kk < 5; kk++) {
    int k0 = kk * 4 + kh;
    v2f bm, a;
    bm.x = hb[k0 * 16 + col];
    bm.y = hb[(k0 + 1) * 16 + col];
    a.x = sW2[col * 20 + k0];
    a.y = sW2[col * 20 + k0 + 1];
    acc2 = wmma4(a, bm, acc2);
  }

  // ---- stage 3: proj = Wproj[n](10x256) @ xp ; column scale (2-att) after ----
  v8f acc3 = {};
  const float* xpp = xp + (b * 256) * HW + hw0 + col;
#pragma unroll 8
  for (int kk = 0; kk < 64; kk++) {
    int k0 = kk * 4 + kh;
    v2f bm, a;
    bm.x = xpp[k0 * HW];
    bm.y = xpp[(k0 + 1) * HW];
    a.x = sWp[col * 256 + k0];
    a.y = sWp[col * 256 + k0 + 1];
    acc3 = wmma4(a, bm, acc3);
  }

  float am = 2.0f - fdep[(n * NB + b) * HW + hw0 + col];
  float* zg = Zg + ((n * NB + b) * 20) * HW + hw0 + col;
  float* zc = Zc + ((n * NB + b) * 20) * HW + hw0 + col;
#pragma unroll
  for (int g = 0; g < 8; g++) {
    int M = mb + g;                          // <= 15; scale arrays padded to 16
    float y2 = fmaxf(acc2[g] * sS2[M] + sT2[M], 0.0f);
    float y3 = fmaxf(acc3[g] * am * sPs[M] + sPt[M], 0.0f);
    float msg = y2 + y3;
    if (M < 10) {                            // predicated stores only
      zg[M * HW] = msg;
      zc[M * HW] = msg;
    }
  }
}

// ---------------------------------------------------------------------------
// K3: gates = sigmoid(conv3x3(Zg, Wg[n]) + bg[n]); writes r*h to Zc ch 10..19.
//     Boundary taps: clamped address + value select (no masked loads).
// ---------------------------------------------------------------------------
__global__ __launch_bounds__(128) void k3_gates(
    const float* __restrict__ Zg, const float* __restrict__ Wg,
    const float* __restrict__ bg, float* __restrict__ G, float* __restrict__ Zc)
{
  __shared__ float sW[5760];  // Wg[n] padded to [32,20,3,3]
  __shared__ float sBg[32];
  int tid = threadIdx.x;
  int wid = blockIdx.x * 4 + (tid >> 5);
  int n = wid >> 12;
  const float* w = Wg + n * 3600;
  for (int i = tid; i < 3600; i += 128) sW[i] = w[i];
  for (int i = 3600 + tid; i < 5760; i += 128) sW[i] = 0.0f;
  if (tid < 32) sBg[tid] = (tid < 20) ? bg[n * 20 + tid] : 0.0f;
  __syncthreads();

  int lane = tid & 31;
  int r = wid & 4095;
  int b = r >> 10;
  int t = r & 1023;
  int ty = t >> 3, x0 = (t & 7) * 16;
  int col = lane & 15, kh = (lane >> 4) << 1;
  int r1 = col + 16;
  const float* zin = Zg + ((n * NB + b) * 20) * HW;

  v8f acc0 = {}, acc1 = {};
#pragma unroll
  for (int tap = 0; tap < 9; tap++) {
    int dy = tap / 3 - 1, dx = tap % 3 - 1;
    int y = ty + dy;
    int x = x0 + col + dx;
    bool valid = ((unsigned)y < 128u) && ((unsigned)x < 128u);
    const float* zp = zin + clampi(y, 0, 127) * 128 + clampi(x, 0, 127);
#pragma unroll
    for (int kk = 0; kk < 5; kk++) {
      int k0 = kk * 4 + kh;
      float v0 = zp[k0 * HW];
      float v1 = zp[(k0 + 1) * HW];
      v2f bm, a0, a1;
      bm.x = valid ? v0 : 0.0f;
      bm.y = valid ? v1 : 0.0f;
      int wi0 = ((col * 20 + k0) * 3 + (dy + 1)) * 3 + (dx + 1);
      int wi1 = ((r1 * 20 + k0) * 3 + (dy + 1)) * 3 + (dx + 1);
      a0.x = sW[wi0]; a0.y = sW[wi0 + 9];
      a1.x = sW[wi1]; a1.y = sW[wi1 + 9];
      acc0 = wmma4(a0, bm, acc0);
      acc1 = wmma4(a1, bm, acc1);
    }
  }

  int hw = ty * 128 + x0 + col;
  float* gout = G + ((n * NB + b) * 20) * HW + hw;
  float* zc = Zc + ((n * NB + b) * 20) * HW + hw;
  int mb = (lane < 16) ? 0 : 8;
#pragma unroll
  for (int g = 0; g < 8; g++) {
    int o = mb + g;                          // <= 15, always a real channel
    float gate = sigm(acc0[g] + sBg[o]);
    int oc = (o < 10) ? o : 9;               // clamp for the unconditional load
    float rp = gate * zin[(10 + oc) * HW + hw];
    gout[o * HW] = gate;
    if (o < 10) zc[(10 + o) * HW] = rp;      // r * h
  }
#pragma unroll
  for (int g = 0; g < 8; g++) {
    int o = 16 + mb + g;                     // sBg padded to 32
    float gate = sigm(acc1[g] + sBg[o]);
    if (o < 20) gout[o * HW] = gate;
  }
}

// ---------------------------------------------------------------------------
// K4: cand = tanh(conv3x3(Zc, Wc[n]) + bc[n]);  xp_new = (1-u)*h + u*cand
// ---------------------------------------------------------------------------
__global__ __launch_bounds__(128) void k4_update(
    const float* __restrict__ Zc, const float* __restrict__ Wc,
    const float* __restrict__ bc, const float* __restrict__ G,
    const float* __restrict__ p_nodes, float* __restrict__ out_xp)
{
  __shared__ float sW[2880];  // Wc[n] padded to [16,20,3,3]
  __shared__ float sBc[16];
  int tid = threadIdx.x;
  int wid = blockIdx.x * 4 + (tid >> 5);
  int n = wid >> 12;
  const float* w = Wc + n * 1800;
  for (int i = tid; i < 1800; i += 128) sW[i] = w[i];
  for (int i = 1800 + tid; i < 2880; i += 128) sW[i] = 0.0f;
  if (tid < 16) sBc[tid] = (tid < 10) ? bc[n * 10 + tid] : 0.0f;
  __syncthreads();

  int lane = tid & 31;
  int r = wid & 4095;
  int b = r >> 10;
  int t = r & 1023;
  int ty = t >> 3, x0 = (t & 7) * 16;
  int col = lane & 15, kh = (lane >> 4) << 1;
  const float* zin = Zc + ((n * NB + b) * 20) * HW;

  v8f acc = {};
#pragma unroll
  for (int tap = 0; tap < 9; tap++) {
    int dy = tap / 3 - 1, dx = tap % 3 - 1;
    int y = ty + dy;
    int x = x0 + col + dx;
    bool valid = ((unsigned)y < 128u) && ((unsigned)x < 128u);
    const float* zp = zin + clampi(y, 0, 127) * 128 + clampi(x, 0, 127);
#pragma unroll
    for (int kk = 0; kk < 5; kk++) {
      int k0 = kk * 4 + kh;
      float v0 = zp[k0 * HW];
      float v1 = zp[(k0 + 1) * HW];
      v2f bm, a;
      bm.x = valid ? v0 : 0.0f;
      bm.y = valid ? v1 : 0.0f;
      int wi = ((col * 20 + k0) * 3 + (dy + 1)) * 3 + (dx + 1);
      a.x = sW[wi]; a.y = sW[wi + 9];
      acc = wmma4(a, bm, acc);
    }
  }

  int hw = ty * 128 + x0 + col;
  int mb = (lane < 16) ? 0 : 8;
#pragma unroll
  for (int g = 0; g < 8; g++) {
    int o = mb + g;                          // <= 15
    int oc = (o < 10) ? o : 9;               // clamp for unconditional loads
    float cand = tanhf(acc[g] + sBc[o]);
    float u = G[((n * NB + b) * 20 + 10 + oc) * HW + hw];
    float h = p_nodes[((n * NB + b) * 10 + oc) * HW + hw];
    float res = (1.0f - u) * h + u * cand;
    if (o < 10) out_xp[((n * NB + b) * 10 + o) * HW + hw] = res;
  }
}

// ---------------------------------------------------------------------------
extern "C" void kernel_launch(void* const* d_in, const int* in_sizes, int n_in,
                              void* d_out, int out_size, void* d_ws, size_t ws_size,
                              hipStream_t stream) {
  (void)in_sizes; (void)n_in; (void)out_size; (void)ws_size;
  const float* h_nodes = (const float*)d_in[1];
  const float* p_nodes = (const float*)d_in[2];
  const float* xp      = (const float*)d_in[3];
  const float* W_dau = (const float*)d_in[4];
  const float* b_dau = (const float*)d_in[5];
  const float* W_dal = (const float*)d_in[6];
  const float* b_dal = (const float*)d_in[7];
  const float* W_dec1 = (const float*)d_in[8];
  const float* dec1_s = (const float*)d_in[9];
  const float* dec1_t = (const float*)d_in[10];
  const float* W_dec2 = (const float*)d_in[11];
  const float* dec2_s = (const float*)d_in[12];
  const float* dec2_t = (const float*)d_in[13];
  const float* Wa0 = (const float*)d_in[14];
  const float* Wa1 = (const float*)d_in[15];
  const float* Wa2 = (const float*)d_in[16];
  const float* Wa3 = (const float*)d_in[17];
  const float* Wa4 = (const float*)d_in[18];
  const float* Wa5 = (const float*)d_in[19];
  const float* b_att = (const float*)d_in[20];
  const float* W_proj = (const float*)d_in[21];
  const float* proj_s = (const float*)d_in[22];
  const float* proj_t = (const float*)d_in[23];
  const float* Wg = (const float*)d_in[24];
  const float* bg = (const float*)d_in[25];
  const float* Wc = (const float*)d_in[26];
  const float* bc = (const float*)d_in[27];

  // workspace layout (floats)
  float* ws = (float*)d_ws;
  float* PA = ws;                         // [6,4,HW]
  float* Zg = PA + 6 * NB * HW;           // [6,4,20,HW]  cat(msg, p)
  float* Zc = Zg + 6 * NB * 20 * HW;      // [6,4,20,HW]  cat(msg, r*p)
  float* G  = Zc + 6 * NB * 20 * HW;      // [6,4,20,HW]  sigmoid gates

  // output layout (floats, concatenated return tuple)
  float* out      = (float*)d_out;
  float* out_xp   = out;                       // [6,4,10,HW]
  float* out_dmu  = out_xp + 6 * NB * 10 * HW; // [4,5,HW]
  float* out_dml  = out_dmu + NB * 5 * HW;     // [4,3,HW]
  float* out_fdep = out_dml + NB * 3 * HW;     // [6,4,HW]

  k1_attention<<<(NB * HW) / 256, 256, 0, stream>>>(
      h_nodes, p_nodes, W_dau, b_dau, W_dal, b_dal,
      Wa0, Wa1, Wa2, Wa3, Wa4, Wa5, b_att,
      out_dmu, out_dml, out_fdep, PA, Zg);

  k2_message<<<(6 * 4096) / 4, 128, 0, stream>>>(
      h_nodes, p_nodes, xp,
      W_dec1, dec1_s, dec1_t, W_dec2, dec2_s, dec2_t,
      W_proj, proj_s, proj_t, out_fdep, PA, Zg, Zc);

  k3_gates<<<(6 * 4096) / 4, 128, 0, stream>>>(Zg, Wg, bg, G, Zc);

  k4_update<<<(6 * 4096) / 4, 128, 0, stream>>>(Zc, Wc, bc, G, p_nodes, out_xp);
}